// ImgtoClass_Metric_14826227106300
// MI455X (gfx1250) — compile-verified
//
#include <hip/hip_runtime.h>

typedef __attribute__((ext_vector_type(2))) float v2f;
typedef __attribute__((ext_vector_type(8))) float v8f;

#define C_DIM   4096
#define B_ROWS  16384
#define KCHUNKS (C_DIM / 4)          // 1024 K-chunks of 4
#define BBUF_ELEMS (KCHUNKS * 64)    // 65536 floats = 256 KB in d_ws

// ---------------------------------------------------------------------------
// Prep: L2-normalize the 3 support rows of x2 and lay them out in the exact
// per-lane order of the V_WMMA_F32_16X16X4_F32 B operand:
//   Bbuf[kchunk*64 + lane*2 + j] = B(k = 4*kchunk + 2*(lane>>4) + j,
//                                    n = lane & 15)
// so the main loop fetches B with a single coalesced float2 load per lane.
// Columns n >= 3 are zero.
// ---------------------------------------------------------------------------
__global__ void prep_kernel(const float* __restrict__ x2,
                            float* __restrict__ Bbuf) {
  __shared__ float red[256];
  __shared__ float rinv[3];
  const int t = threadIdx.x;

  for (int n = 0; n < 3; ++n) {
    float ss = 0.f;
    for (int k = t; k < C_DIM; k += 256) {
      float v = x2[n * C_DIM + k];
      ss = fmaf(v, v, ss);
    }
    red[t] = ss;
    __syncthreads();
    for (int off = 128; off > 0; off >>= 1) {
      if (t < off) red[t] += red[t + off];
      __syncthreads();
    }
    if (t == 0) rinv[n] = rsqrtf(red[0]);
    __syncthreads();
  }

  for (int idx = t; idx < BBUF_ELEMS; idx += 256) {
    const int kchunk = idx >> 6;
    const int rem    = idx & 63;
    const int lane   = rem >> 1;
    const int j      = rem & 1;
    const int n      = lane & 15;
    const int k      = 4 * kchunk + 2 * (lane >> 4) + j;
    float v = 0.f;
    if (n < 3) v = x2[n * C_DIM + k] * rinv[n];
    Bbuf[idx] = v;
  }
}

// ---------------------------------------------------------------------------
// Main: one wave32 owns 16 rows of x1. Streams K in chunks of 4 through
// V_WMMA_F32_16X16X4_F32 (A = 16 query rows, B = 3 normalized supports),
// while accumulating sum-of-squares per row with co-executing VALU FMAs.
// Final: out[row][n] = dot / ||x1_row||.
// ---------------------------------------------------------------------------
__global__ void __launch_bounds__(256)
cosine_topk_kernel(const float* __restrict__ x1,
                   const float* __restrict__ Bbuf,
                   float* __restrict__ out) {
  const int lane    = threadIdx.x & 31;
  const int wave    = threadIdx.x >> 5;
  const int rowbase = (blockIdx.x * 8 + wave) * 16;
  const int m16     = lane & 15;   // row within tile (also N column for D)
  const int hi      = lane >> 4;   // 0: K {0,1} / 1: K {2,3} within chunk

  // A operand: lane L<16 -> row L, K kb+{0,1}; lane L+16 -> row L, K kb+{2,3}
  const float* aptr = x1 + (size_t)(rowbase + m16) * C_DIM + 2 * hi;
  // B operand stream, pre-swizzled by prep_kernel
  const float* bptr = Bbuf + 2 * lane;

  v8f acc = {0.f, 0.f, 0.f, 0.f, 0.f, 0.f, 0.f, 0.f};
  float ss = 0.f;

  for (int ko = 0; ko < KCHUNKS; ko += 8) {
    // one cache line ahead on the hot x1 stream -> global_prefetch_b8
    __builtin_prefetch(aptr + 64, 0, 3);
#pragma unroll
    for (int u = 0; u < 8; ++u) {
      v2f a = *(const v2f*)aptr;           // global_load_b64 (x1)
      v2f b = *(const v2f*)bptr;           // global_load_b64 (Bbuf, L2-hot)
      ss = fmaf(a.x, a.x, ss);
      ss = fmaf(a.y, a.y, ss);
      // D = A(16x4) * B(4x16) + C   (fp32 WMMA, wave32)
      acc = __builtin_amdgcn_wmma_f32_16x16x4_f32(
          /*neg_a=*/false, a, /*neg_b=*/false, b,
          /*c_mod=*/(short)0, acc, /*reuse_a=*/false, /*reuse_b=*/false);
      aptr += 4;
      bptr += 64;
    }
  }

  // Row sum-of-squares: lane L holds K%4 in {0,1}, lane L+16 holds {2,3}.
  ss += __shfl_xor(ss, 16, 32);            // now lane L and L+16 hold row L's sumsq

  // D layout: VGPR v, lanes 0-15 -> (M=v,   N=lane),
  //                   lanes 16-31 -> (M=v+8, N=lane-16). Emit N < 3.
#pragma unroll
  for (int v = 0; v < 8; ++v) {
    const int m = v + (hi << 3);
    const float rn = __shfl(ss, m, 32);    // row m's sum-of-squares
    if (m16 < 3) {
      out[(size_t)(rowbase + m) * 3 + m16] = acc[v] * rsqrtf(rn);
    }
  }
}

// ---------------------------------------------------------------------------
extern "C" void kernel_launch(void* const* d_in, const int* in_sizes, int n_in,
                              void* d_out, int out_size, void* d_ws, size_t ws_size,
                              hipStream_t stream) {
  (void)in_sizes; (void)n_in; (void)out_size; (void)ws_size;
  const float* x1 = (const float*)d_in[0];   // [16384, 4096] fp32
  const float* x2 = (const float*)d_in[1];   // [3, 4096] fp32
  float* out  = (float*)d_out;               // [16384, 3] fp32
  float* Bbuf = (float*)d_ws;                // 256 KB WMMA-B staging

  prep_kernel<<<1, 256, 0, stream>>>(x2, Bbuf);
  cosine_topk_kernel<<<B_ROWS / 128, 256, 0, stream>>>(x1, Bbuf, out);
}